// GCN_50886772523112
// MI455X (gfx1250) — compile-verified
//
#include <hip/hip_runtime.h>

#define HIDDEN 128
#define NODES 64
#define EPS 1e-5f

typedef __attribute__((ext_vector_type(2))) float v2f;
typedef __attribute__((ext_vector_type(8))) float v8f;

// ---------------- embedding: h = class_emb[x0] + neigh_emb[x1] ----------------
__global__ __launch_bounds__(256) void k_embed(const int* __restrict__ x,
    const float* __restrict__ ce, const float* __restrict__ ne,
    float* __restrict__ h, int N) {
  int t = blockIdx.x * 256 + threadIdx.x;
  int n = t >> 5;
  if (n >= N) return;
  int k4 = (t & 31) * 4;
  int c0 = x[2 * n + 0], c1 = x[2 * n + 1];
  float4 a = *(const float4*)(ce + (size_t)c0 * HIDDEN + k4);
  float4 b = *(const float4*)(ne + (size_t)c1 * HIDDEN + k4);
  float4 r;
  r.x = a.x + b.x; r.y = a.y + b.y; r.z = a.z + b.z; r.w = a.w + b.w;
  *(float4*)(h + (size_t)n * HIDDEN + k4) = r;
}

// ---------------- degree / dinv (layers 0..2 share edges) ----------------
__global__ __launch_bounds__(256) void k_deg_init(float* __restrict__ deg, int N) {
  int i = blockIdx.x * 256 + threadIdx.x;
  if (i < N) deg[i] = 1.0f;  // self loop
}
__global__ __launch_bounds__(256) void k_deg_count(const int* __restrict__ dst,
                                                   float* __restrict__ deg, int E) {
  int e = blockIdx.x * 256 + threadIdx.x;
  if (e < E) atomicAdd(&deg[dst[e]], 1.0f);
}
__global__ __launch_bounds__(256) void k_dinv(const float* __restrict__ deg,
                                              float* __restrict__ dinv, int N) {
  int i = blockIdx.x * 256 + threadIdx.x;
  if (i < N) dinv[i] = rsqrtf(deg[i]);
}

// ---------------- hw = h @ W via V_WMMA_F32_16X16X4_F32 ----------------
// One wave per 16-row strip; 8 accumulators cover all 128 output columns.
// W staged in LDS PAIR-INTERLEAVED: Wl[k/2][c] = {W[k][c], W[k+1][c]} so each
// B fragment is a single aligned ds_load_b64 into an even VGPR pair (no repack
// moves between DS loads and WMMA operands).
__global__ __launch_bounds__(256) void k_gemm(const float* __restrict__ h,
    const float* __restrict__ W, float* __restrict__ hw, int N) {
  __shared__ v2f Wl[(HIDDEN / 2) * HIDDEN];  // 64 KB
  for (int i = threadIdx.x; i < (HIDDEN / 2) * HIDDEN; i += 256) {
    int kp = i >> 7;     // K pair index
    int c  = i & 127;    // column
    v2f p;
    p.x = W[(size_t)(2 * kp + 0) * HIDDEN + c];
    p.y = W[(size_t)(2 * kp + 1) * HIDDEN + c];
    Wl[i] = p;
  }
  __syncthreads();

  int lane = threadIdx.x & 31, wave = threadIdx.x >> 5;
  int m0 = (blockIdx.x * 8 + wave) * 16;
  if (m0 >= N) return;
  int half = lane >> 4;      // 0: K pair (k,k+1), 1: K pair (k+2,k+3)
  int l15 = lane & 15;

  v8f acc[8] = {};
  const float* arow = h + (size_t)(m0 + l15) * HIDDEN + 2 * half;
  __builtin_prefetch(arow + 64, 0, 0);   // keep A stream ahead of the k-loop

  #pragma unroll 4
  for (int k = 0; k < HIDDEN; k += 4) {
    // A frag (16x4 f32): lanes 0-15 hold K=k,k+1 for M=lane&15; lanes 16-31 K=k+2,k+3
    v2f a;
    a.x = arow[k];
    a.y = arow[k + 1];
    const v2f* w0 = Wl + ((k >> 1) + half) * HIDDEN + l15;  // rows k'(=k+2h), k'+1
    #pragma unroll
    for (int t = 0; t < 8; ++t) {
      v2f b = w0[t * 16];  // single ds_load_b64, already in WMMA operand layout
      acc[t] = __builtin_amdgcn_wmma_f32_16x16x4_f32(
          false, a, false, b, (short)0, acc[t], false, false);
    }
  }

  // D 16x16 f32 layout: vgpr i -> row m0 + i + 8*half, col n0 + l15
  float* obase = hw + (size_t)(m0 + 8 * half) * HIDDEN + l15;
  #pragma unroll
  for (int t = 0; t < 8; ++t) {
    #pragma unroll
    for (int i = 0; i < 8; ++i)
      obase[(size_t)i * HIDDEN + t * 16] = acc[t][i];
  }
}

// ---------------- agg = hw*dinv^2 + bias (self loop + bias) ----------------
__global__ __launch_bounds__(256) void k_agg_init(const float* __restrict__ hw,
    const float* __restrict__ dinv, const float* __restrict__ bias,
    float* __restrict__ agg, int N) {
  int t = blockIdx.x * 256 + threadIdx.x;
  int n = t >> 5;
  if (n >= N) return;
  int k4 = (t & 31) * 4;
  float d = dinv[n], d2 = d * d;
  float4 v = *(const float4*)(hw + (size_t)n * HIDDEN + k4);
  float4 b = *(const float4*)(bias + k4);
  float4 r;
  r.x = v.x * d2 + b.x; r.y = v.y * d2 + b.y;
  r.z = v.z * d2 + b.z; r.w = v.w * d2 + b.w;
  *(float4*)(agg + (size_t)n * HIDDEN + k4) = r;
}

// ---------------- edge scatter: agg[dst] += hw[src]*dinv[src]*dinv[dst] ----------------
// One wave per edge; uniform src/dst loads scalarize; float4 gather + 4 atomics/lane.
__global__ __launch_bounds__(256) void k_scatter(const int* __restrict__ src,
    const int* __restrict__ dst, const float* __restrict__ hw,
    const float* __restrict__ dinv, float* __restrict__ agg, int E) {
  int t = blockIdx.x * 256 + threadIdx.x;
  int e = t >> 5;
  if (e >= E) return;
  int k4 = (t & 31) * 4;
  int s = src[e], d = dst[e];
  float w = dinv[s] * dinv[d];
  float4 v = *(const float4*)(hw + (size_t)s * HIDDEN + k4);
  float* o = agg + (size_t)d * HIDDEN + k4;
  atomicAdd(o + 0, v.x * w);
  atomicAdd(o + 1, v.y * w);
  atomicAdd(o + 2, v.z * w);
  atomicAdd(o + 3, v.w * w);
}

// ---------------- wave32 reduction ----------------
__device__ __forceinline__ float wave_sum32(float s) {
  #pragma unroll
  for (int off = 16; off > 0; off >>= 1) s += __shfl_xor(s, off, 32);
  return s;
}

// ---------------- h = LN(h + relu(agg)) ----------------
__global__ __launch_bounds__(256) void k_post_ln(float* __restrict__ h,
    const float* __restrict__ agg, const float* __restrict__ g,
    const float* __restrict__ bln, int N) {
  int node = blockIdx.x * 8 + (threadIdx.x >> 5);
  if (node >= N) return;
  int k4 = (threadIdx.x & 31) * 4;
  float4 hv = *(const float4*)(h + (size_t)node * HIDDEN + k4);
  float4 av = *(const float4*)(agg + (size_t)node * HIDDEN + k4);
  float4 r;
  r.x = hv.x + fmaxf(av.x, 0.f);
  r.y = hv.y + fmaxf(av.y, 0.f);
  r.z = hv.z + fmaxf(av.z, 0.f);
  r.w = hv.w + fmaxf(av.w, 0.f);
  float mu = wave_sum32(r.x + r.y + r.z + r.w) * (1.f / HIDDEN);
  float dx = r.x - mu, dy = r.y - mu, dz = r.z - mu, dw = r.w - mu;
  float var = wave_sum32(dx * dx + dy * dy + dz * dz + dw * dw) * (1.f / HIDDEN);
  float inv = rsqrtf(var + EPS);
  float4 gv = *(const float4*)(g + k4);
  float4 bv = *(const float4*)(bln + k4);
  float4 o;
  o.x = dx * inv * gv.x + bv.x;
  o.y = dy * inv * gv.y + bv.y;
  o.z = dz * inv * gv.z + bv.z;
  o.w = dw * inv * gv.w + bv.w;
  *(float4*)(h + (size_t)node * HIDDEN + k4) = o;
}

// ---------------- layer 3: per-graph column sums S[g] = sum_i hw[i] ----------------
__global__ __launch_bounds__(128) void k_graph_sum(const float* __restrict__ hw,
                                                   float* __restrict__ S) {
  int g = blockIdx.x, k = threadIdx.x;
  const float* base = hw + (size_t)g * NODES * HIDDEN + k;
  float s = 0.f;
  #pragma unroll 8
  for (int i = 0; i < NODES; ++i) s += base[(size_t)i * HIDDEN];
  S[(size_t)g * HIDDEN + k] = s;
}

// ---------------- layer 3 post: h = LN(h + relu((S+hw)/65 + b)) ----------------
__global__ __launch_bounds__(256) void k_post_fa(float* __restrict__ h,
    const float* __restrict__ hw, const float* __restrict__ S,
    const float* __restrict__ bias, const float* __restrict__ g,
    const float* __restrict__ bln, int N) {
  int node = blockIdx.x * 8 + (threadIdx.x >> 5);
  if (node >= N) return;
  int k4 = (threadIdx.x & 31) * 4;
  int gr = node >> 6;  // / NODES
  const float inv65 = 1.0f / 65.0f;  // deg = 64 FA edges + 1 self loop
  float4 sv = *(const float4*)(S + (size_t)gr * HIDDEN + k4);
  float4 wv = *(const float4*)(hw + (size_t)node * HIDDEN + k4);
  float4 bb = *(const float4*)(bias + k4);
  float4 hv = *(const float4*)(h + (size_t)node * HIDDEN + k4);
  float4 r;
  r.x = hv.x + fmaxf((sv.x + wv.x) * inv65 + bb.x, 0.f);
  r.y = hv.y + fmaxf((sv.y + wv.y) * inv65 + bb.y, 0.f);
  r.z = hv.z + fmaxf((sv.z + wv.z) * inv65 + bb.z, 0.f);
  r.w = hv.w + fmaxf((sv.w + wv.w) * inv65 + bb.w, 0.f);
  float mu = wave_sum32(r.x + r.y + r.z + r.w) * (1.f / HIDDEN);
  float dx = r.x - mu, dy = r.y - mu, dz = r.z - mu, dw = r.w - mu;
  float var = wave_sum32(dx * dx + dy * dy + dz * dz + dw * dw) * (1.f / HIDDEN);
  float inv = rsqrtf(var + EPS);
  float4 gv = *(const float4*)(g + k4);
  float4 bv = *(const float4*)(bln + k4);
  float4 o;
  o.x = dx * inv * gv.x + bv.x;
  o.y = dy * inv * gv.y + bv.y;
  o.z = dz * inv * gv.z + bv.z;
  o.w = dw * inv * gv.w + bv.w;
  *(float4*)(h + (size_t)node * HIDDEN + k4) = o;
}

// ---------------- out[g] = h[g*64] @ out_W + out_b ----------------
__global__ __launch_bounds__(256) void k_out(const float* __restrict__ h,
    const float* __restrict__ oW, const float* __restrict__ ob,
    float* __restrict__ out, int G) {
  int g = blockIdx.x * 8 + (threadIdx.x >> 5);
  if (g >= G) return;
  int c = threadIdx.x & 31;
  const float* row = h + (size_t)g * NODES * HIDDEN;  // root node of graph g
  float s = ob[c];
  #pragma unroll 4
  for (int k = 0; k < HIDDEN; ++k) s = fmaf(row[k], oW[(size_t)k * 32 + c], s);
  out[(size_t)g * 32 + c] = s;
}

extern "C" void kernel_launch(void* const* d_in, const int* in_sizes, int n_in,
                              void* d_out, int out_size, void* d_ws, size_t ws_size,
                              hipStream_t stream) {
  const int N = in_sizes[0] / 2;     // 16384
  const int E = in_sizes[1] / 2;     // 262144
  const int G = N / NODES;           // 256

  const int*   x   = (const int*)d_in[0];
  const int*   src = (const int*)d_in[1];
  const int*   dst = src + E;
  const float* ce  = (const float*)d_in[3];
  const float* ne  = (const float*)d_in[4];
  const float* Ws  = (const float*)d_in[5];
  const float* bs  = (const float*)d_in[6];
  const float* lng = (const float*)d_in[7];
  const float* lnb = (const float*)d_in[8];
  const float* oW  = (const float*)d_in[9];
  const float* ob  = (const float*)d_in[10];
  float* out = (float*)d_out;

  float* h    = (float*)d_ws;                 // N*128
  float* hw   = h   + (size_t)N * HIDDEN;     // N*128
  float* agg  = hw  + (size_t)N * HIDDEN;     // N*128
  float* deg  = agg + (size_t)N * HIDDEN;     // N
  float* dinv = deg + N;                      // N
  float* S    = dinv + N;                     // G*128

  dim3 blk(256);
  int nwave_blocks = (N * 32 + 255) / 256;    // one wave per node, 4 floats/lane
  int gemm_blocks  = (N / 16 + 7) / 8;        // one wave per 16-row strip

  k_embed<<<nwave_blocks, blk, 0, stream>>>(x, ce, ne, h, N);

  // degrees shared by layers 0..2
  k_deg_init<<<(N + 255) / 256, blk, 0, stream>>>(deg, N);
  k_deg_count<<<(E + 255) / 256, blk, 0, stream>>>(dst, deg, E);
  k_dinv<<<(N + 255) / 256, blk, 0, stream>>>(deg, dinv, N);

  for (int l = 0; l < 3; ++l) {
    k_gemm<<<gemm_blocks, blk, 0, stream>>>(h, Ws + (size_t)l * HIDDEN * HIDDEN, hw, N);
    k_agg_init<<<nwave_blocks, blk, 0, stream>>>(hw, dinv, bs + l * HIDDEN, agg, N);
    k_scatter<<<(E * 32 + 255) / 256, blk, 0, stream>>>(src, dst, hw, dinv, agg, E);
    k_post_ln<<<(N + 7) / 8, blk, 0, stream>>>(h, agg, lng + l * HIDDEN, lnb + l * HIDDEN, N);
  }

  // layer 3: fully-adjacent aggregation collapses to per-graph sum
  k_gemm<<<gemm_blocks, blk, 0, stream>>>(h, Ws + (size_t)3 * HIDDEN * HIDDEN, hw, N);
  k_graph_sum<<<G, dim3(128), 0, stream>>>(hw, S);
  k_post_fa<<<(N + 7) / 8, blk, 0, stream>>>(h, hw, S, bs + 3 * HIDDEN,
                                             lng + 3 * HIDDEN, lnb + 3 * HIDDEN, N);

  k_out<<<(G + 7) / 8, blk, 0, stream>>>(h, oW, ob, out, G);
}